// OrientationDetector_25056839205935
// MI455X (gfx1250) — compile-verified
//
#include <hip/hip_runtime.h>

#define PS 32
#define NBINS 36
#define WAVES_PER_BLOCK 4
#define PATCHES_PER_WAVE 4

#ifndef __has_builtin
#define __has_builtin(x) 0
#endif

#if __has_builtin(__builtin_amdgcn_global_load_async_to_lds_b128)
#define HAS_ASYNC_LDS 1
#else
#define HAS_ASYNC_LDS 0
#endif

#if __has_builtin(__builtin_amdgcn_s_wait_asynccnt)
#define WAIT_ASYNC(n) do { __builtin_amdgcn_s_wait_asynccnt(n); asm volatile("" ::: "memory"); } while (0)
#else
#define WAIT_ASYNC(n) asm volatile("s_wait_asynccnt %0" :: "i"(n) : "memory")
#endif

#define AS1 __attribute__((address_space(1)))
#define AS3 __attribute__((address_space(3)))

typedef int v4i __attribute__((vector_size(16)));

#if HAS_ASYNC_LDS
// DMA one 4KB patch (32x32 f32) into a per-wave LDS buffer.
// Each lane moves 16B per issue; 8 issues x 32 lanes x 16B = 4096B.
__device__ __forceinline__ void issue_patch_load(const float* gsrc, float* lbuf, int lane) {
  AS1 v4i* g = (AS1 v4i*)((AS1 char*)gsrc + lane * 16);
  AS3 v4i* l = (AS3 v4i*)((AS3 char*)lbuf + lane * 16);
  __builtin_amdgcn_global_load_async_to_lds_b128(g, l, 0,    0);
  __builtin_amdgcn_global_load_async_to_lds_b128(g, l, 512,  0);
  __builtin_amdgcn_global_load_async_to_lds_b128(g, l, 1024, 0);
  __builtin_amdgcn_global_load_async_to_lds_b128(g, l, 1536, 0);
  __builtin_amdgcn_global_load_async_to_lds_b128(g, l, 2048, 0);
  __builtin_amdgcn_global_load_async_to_lds_b128(g, l, 2560, 0);
  __builtin_amdgcn_global_load_async_to_lds_b128(g, l, 3072, 0);
  __builtin_amdgcn_global_load_async_to_lds_b128(g, l, 3584, 0);
}
#endif

// Minimax atan2: degree-9 odd polynomial on [0,1], max err ~1e-5 rad
// (orientation feeds 10-degree-wide bins, so this is far below observability).
// Matches numpy conventions incl. atan2(0,0)=0 and the x<0 half-plane.
__device__ __forceinline__ float fast_atan2f(float yy, float xx) {
  const float PI_F   = 3.14159265358979f;
  const float PIO2_F = 1.57079632679490f;
  float ax = __builtin_fabsf(xx);
  float ay = __builtin_fabsf(yy);
  float mx = fmaxf(ax, ay);
  float mn = fminf(ax, ay);
  float r  = __builtin_amdgcn_rcpf(mx);          // v_rcp_f32 (~1 ulp)
  float t  = (mx > 0.0f) ? mn * r : 0.0f;        // guard atan2(0,0)
  float s  = t * t;
  float p  = fmaf(s, 0.0208351f, -0.0851330f);
  p = fmaf(s, p, 0.1801410f);
  p = fmaf(s, p, -0.3302995f);
  p = fmaf(s, p, 0.9998660f);
  float a  = p * t;
  if (ay > ax)   a = PIO2_F - a;
  if (xx < 0.0f) a = PI_F - a;
  return copysignf(a, yy);
}

// One wave (32 lanes) processes one 32x32 patch. lane = column x.
// x-neighbors via clamped LDS loads (patch is resident in LDS), y-neighbors
// via a rolling register window, histogram via per-wave ds_add_f32 atomics.
__device__ __forceinline__ void process_patch(const float* __restrict__ src,
                                              int lane, float exv, float gsc,
                                              float* __restrict__ hist,
                                              float* __restrict__ outp) {
  const float PI_F = 3.14159265358979323846f;

  // zero the 36-bin per-wave histogram (wave-private: no block barrier needed,
  // LDS ops from the same wave are in-order)
  hist[lane] = 0.0f;
  if (lane < NBINS - PS) hist[PS + lane] = 0.0f;
  asm volatile("" ::: "memory");

  const int offL = (lane == 0)      ? 0      : lane - 1;  // replicate-pad x
  const int offR = (lane == PS - 1) ? PS - 1 : lane + 1;

  float curv  = src[lane];   // row 0
  float prevv = curv;        // replicate-pad for y==0
#pragma unroll 4
  for (int y = 0; y < PS; ++y) {
    const int rowN = (y < PS - 1) ? y + 1 : PS - 1;       // clamped: row31 == curv
    float nextv = src[rowN * PS + lane];
    float lft   = src[y * PS + offL];
    float rgt   = src[y * PS + offR];
    float gx = 0.5f * (lft - rgt);           // 0.5*(x[i-1] - x[i+1])
    float gy = 0.5f * (prevv - nextv);       // 0.5*(x[y-1] - x[y+1])
    float eyv = __shfl(exv, y);              // uniform index -> v_readlane
    float mag = sqrtf(gx * gx + gy * gy + 1e-10f) * (gsc * exv * eyv);
    float ori = fast_atan2f(gy, gx);
    float o   = (ori + PI_F) * 5.7295779513082320877f;    // 36/(2*pi)
    float b0f = floorf(o);
    float w1  = o - b0f;
    int b0 = (int)b0f;
    if (b0 >= NBINS) b0 -= NBINS;            // ori == +pi wraps to bin 0
    if (b0 < 0)      b0 = 0;                 // safety vs approx rounding
    atomicAdd(&hist[b0], (1.0f - w1) * mag); // ds_add_f32
    prevv = curv;
    curv  = nextv;
  }
  asm volatile("" ::: "memory");

  // lane 0: 3-tap smoothing (zero-padded) + first-occurrence argmax
  if (lane == 0) {
    const float inv = 1.0f / (float)(PS * PS);
    float hm1 = 0.0f;
    float h0  = hist[0] * inv;
    float best = -3.0e38f;
    int bi = 0;
    for (int i = 0; i < NBINS; ++i) {
      float hp1 = (i < NBINS - 1) ? hist[i + 1] * inv : 0.0f;
      float sm = 0.33f * hm1 + 0.34f * h0 + 0.33f * hp1;
      if (sm > best) { best = sm; bi = i; }
      hm1 = h0;
      h0  = hp1;
    }
    *outp = PI_F - (float)bi * 0.17453292519943295769f;  // -(2*pi*bi/36 - pi)
  }
}

__global__ void __launch_bounds__(PS * WAVES_PER_BLOCK)
OrientationDetector_kernel(const float* __restrict__ x, float* __restrict__ out, int B) {
  const int lane   = threadIdx.x & (PS - 1);
  const int w      = threadIdx.x >> 5;
  const int gw     = blockIdx.x * WAVES_PER_BLOCK + w;
  const int stride = gridDim.x * WAVES_PER_BLOCK;

  // Separable circular Gaussian: gk(x,y) = 10 * e(x)*e(y) / (sum e)^2,
  // e(t) = exp(-t^2 / (0.9*16^2)), t = linspace(-16,16,32)
  float xl  = -16.0f + (float)lane * (32.0f / 31.0f);
  float exv = expf(-(xl * xl) * (1.0f / 230.4f));
  float s = exv;
#pragma unroll
  for (int off = 16; off > 0; off >>= 1) s += __shfl_xor(s, off);
  float gsc = 10.0f / (s * s);

  __shared__ float s_hist[WAVES_PER_BLOCK][NBINS + 4];

#if HAS_ASYNC_LDS
  __shared__ float s_buf[2][WAVES_PER_BLOCK][PS * PS];

  int patch = gw;
  if (patch < B) issue_patch_load(x + (size_t)patch * (PS * PS), &s_buf[0][w][0], lane);
  int buf = 0;
  for (; patch < B; patch += stride, buf ^= 1) {
    int nxt = patch + stride;
    if (nxt < B) {
      // double-buffer: kick off next patch's DMA, then wait for current's 8
      // issues (async loads complete in order -> cnt<=8 means current is done)
      issue_patch_load(x + (size_t)nxt * (PS * PS), &s_buf[buf ^ 1][w][0], lane);
      WAIT_ASYNC(8);
    } else {
      WAIT_ASYNC(0);
    }
    process_patch(&s_buf[buf][w][0], lane, exv, gsc, &s_hist[w][0], out + patch);
  }
#else
  for (int patch = gw; patch < B; patch += stride) {
    const float* src = x + (size_t)patch * (PS * PS);
    int nxt = patch + stride;
    if (nxt < B) {
      // prefetch next patch (32 lanes x 128B cachelines = 4KB)
      __builtin_prefetch(x + (size_t)nxt * (PS * PS) + lane * PS, 0, 1);
    }
    process_patch(src, lane, exv, gsc, &s_hist[w][0], out + patch);
  }
#endif
}

extern "C" void kernel_launch(void* const* d_in, const int* in_sizes, int n_in,
                              void* d_out, int out_size, void* d_ws, size_t ws_size,
                              hipStream_t stream) {
  (void)n_in; (void)d_ws; (void)ws_size; (void)out_size;
  const float* x = (const float*)d_in[0];
  float* out = (float*)d_out;
  int B = in_sizes[0] / (PS * PS);   // 65536 patches

  int totalWaves = (B + PATCHES_PER_WAVE - 1) / PATCHES_PER_WAVE;
  int blocks = (totalWaves + WAVES_PER_BLOCK - 1) / WAVES_PER_BLOCK;
  if (blocks < 1) blocks = 1;

  OrientationDetector_kernel<<<blocks, PS * WAVES_PER_BLOCK, 0, stream>>>(x, out, B);
}